// GraphAttentionModule_9500467659171
// MI455X (gfx1250) — compile-verified
//
#include <hip/hip_runtime.h>
#include <cstdint>

// GATv2 x3 + output projection for MI455X (gfx1250, wave32).
// Dense node GEMMs use native fp32 WMMA (v_wmma_f32_16x16x4_f32).
// Edge softmax/aggregation uses wave-per-edge float4 gathers + L2 float atomics.

typedef __attribute__((ext_vector_type(2))) float v2f;
typedef __attribute__((ext_vector_type(8))) float v8f;

#define WPB 8  // waves per block (256 threads)

// ---- monotone key punning for float atomicMax over signed floats ----
__device__ __forceinline__ unsigned int fkey(float f) {
  unsigned int u = __float_as_uint(f);
  return (u & 0x80000000u) ? ~u : (u | 0x80000000u);
}
__device__ __forceinline__ float funkey(unsigned int k) {
  unsigned int b = (k & 0x80000000u) ? (k ^ 0x80000000u) : ~k;
  return __uint_as_float(b);
}

__global__ void zero_f32(float* __restrict__ p, long n) {
  long i = (long)blockIdx.x * blockDim.x + threadIdx.x;
  if (i < n) p[i] = 0.0f;
}

// ================= fp32 WMMA GEMM: Y[rows,128] = X[rows,K] @ W[K,128] + b ====
// grid.x = rows/16, block = 256 (8 waves; wave w owns column tile w*16).
// A frag (16x4 f32): lane m = lane&15; VGPR0=K(2*half), VGPR1=K(2*half+1)
// B frag (4x16 f32): N = lane&15;    VGPR0=K(2*half), VGPR1=K(2*half+1)
// C/D (16x16 f32):   VGPR r -> M = r + 8*half, N = lane&15
template <int K>
__global__ void gemm_wmma_f32(const float* __restrict__ X,
                              const float* __restrict__ W,
                              const float* __restrict__ bias,
                              float* __restrict__ Y) {
  __shared__ float Xs[16][K + 4];
  const int tid = threadIdx.x;
  const long row0 = (long)blockIdx.x * 16;
  for (int i = tid; i < 16 * K; i += 256) {
    int r = i / K, c = i - r * K;
    Xs[r][c] = X[(row0 + r) * K + c];
  }
  __syncthreads();
  const int wave = tid >> 5, lane = tid & 31;
  const int col0 = wave * 16;
  const int half = lane >> 4;
  const int l = lane & 15;
  v8f acc = {};
#pragma unroll
  for (int kb = 0; kb < K; kb += 4) {
    const int ka = kb + 2 * half;
    v2f a, b;
    a.x = Xs[l][ka];
    a.y = Xs[l][ka + 1];
    b.x = W[(long)ka * 128 + col0 + l];
    b.y = W[(long)(ka + 1) * 128 + col0 + l];
    acc = __builtin_amdgcn_wmma_f32_16x16x4_f32(false, a, false, b, (short)0,
                                                acc, false, false);
  }
  const float bv = bias[col0 + l];
#pragma unroll
  for (int r = 0; r < 8; ++r)
    Y[(row0 + r + 8 * half) * 128 + col0 + l] = acc[r] + bv;
}

// =================== self loops: mean incoming edge_attr per dst ============
__global__ void selfloop_accum(const int* __restrict__ dst,
                               const float* __restrict__ ea,
                               float* __restrict__ deg,
                               float* __restrict__ easum, int E) {
  long t = (long)blockIdx.x * blockDim.x + threadIdx.x;
  if (t >= E) return;
  int d = dst[t];
  float4 e = *(const float4*)(ea + t * 4);
  atomicAdd(&deg[d], 1.0f);
  float* p = easum + (long)d * 4;
  atomicAdd(p + 0, e.x);
  atomicAdd(p + 1, e.y);
  atomicAdd(p + 2, e.z);
  atomicAdd(p + 3, e.w);
}

__global__ void selfloop_final(float* __restrict__ deg,
                               float* __restrict__ easum, int n) {
  long t = (long)blockIdx.x * blockDim.x + threadIdx.x;
  if (t >= n) return;
  float g = 1.0f / fmaxf(deg[t], 1.0f);
  float* p = easum + t * 4;
  p[0] *= g; p[1] *= g; p[2] *= g; p[3] *= g;  // in place -> loop_ea
}

// =================== pass 1: alpha + segment max (wave per edge) ============
__global__ void edge_alpha(const float* __restrict__ xl,
                           const float* __restrict__ xr,
                           const int* __restrict__ src,
                           const int* __restrict__ dst,
                           const float* __restrict__ ea,
                           const float* __restrict__ loop_ea,
                           const float* __restrict__ We,
                           const float* __restrict__ att,
                           float* __restrict__ alpha,
                           unsigned int* __restrict__ amax_bits,
                           int Eo, int Etot) {
  __shared__ float Wes[512];   // We: [4,128]
  __shared__ float atts[128];  // att flat: j = h*16+c
  const int tid = threadIdx.x;
  for (int i = tid; i < 512; i += 256) Wes[i] = We[i];
  if (tid < 128) atts[tid] = att[tid];
  __syncthreads();
  const long e = (long)blockIdx.x * WPB + (tid >> 5);
  if (e >= Etot) return;
  const int lane = tid & 31;
  int s, d; float4 ev;
  if (e < Eo) {
    s = src[e]; d = dst[e];
    ev = *(const float4*)(ea + e * 4);
  } else {
    s = d = (int)(e - Eo);
    ev = *(const float4*)(loop_ea + (long)s * 4);
  }
  const int j0 = lane * 4;
  float4 xa = *(const float4*)(xl + (long)s * 128 + j0);
  float4 xb = *(const float4*)(xr + (long)d * 128 + j0);
  float p = 0.0f;
#pragma unroll
  for (int i = 0; i < 4; ++i) {
    const int j = j0 + i;
    float ej = ev.x * Wes[j] + ev.y * Wes[128 + j] + ev.z * Wes[256 + j] +
               ev.w * Wes[384 + j];
    float v = ((const float*)&xa)[i] + ((const float*)&xb)[i] + ej;
    v = v > 0.0f ? v : 0.2f * v;  // leaky_relu(0.2)
    p += v * atts[j];
  }
  p += __shfl_xor(p, 1, 32);
  p += __shfl_xor(p, 2, 32);
  if ((lane & 3) == 0) {
    const int h = lane >> 2;
    alpha[e * 8 + h] = p;
    atomicMax(&amax_bits[(long)d * 8 + h], fkey(p));
  }
}

__global__ void decode_amax(unsigned int* __restrict__ p, long n) {
  long i = (long)blockIdx.x * blockDim.x + threadIdx.x;
  if (i < n) {
    unsigned int k = p[i];
    ((float*)p)[i] = funkey(k);
  }
}

// =================== pass 2: exp + segment sum (thread per edge,head) =======
__global__ void edge_exp(float* __restrict__ alpha, const int* __restrict__ dst,
                         const float* __restrict__ amax,
                         float* __restrict__ den, int Eo, long total) {
  long t = (long)blockIdx.x * blockDim.x + threadIdx.x;
  if (t >= total) return;
  long e = t >> 3;
  int h = (int)(t & 7);
  int d = (e < Eo) ? dst[e] : (int)(e - Eo);
  float v = __expf(alpha[t] - amax[(long)d * 8 + h]);
  alpha[t] = v;  // overwrite with exp
  atomicAdd(&den[(long)d * 8 + h], v);
}

// =================== pass 3: weighted scatter-add (wave per edge) ===========
__global__ void edge_scatter(const float* __restrict__ xl,
                             const float* __restrict__ ex,
                             const int* __restrict__ src,
                             const int* __restrict__ dst,
                             const float* __restrict__ den,
                             float* __restrict__ out, int Eo, int Etot) {
  const long e = (long)blockIdx.x * WPB + (threadIdx.x >> 5);
  if (e >= Etot) return;
  const int lane = threadIdx.x & 31;
  int s, d;
  if (e < Eo) { s = src[e]; d = dst[e]; } else { s = d = (int)(e - Eo); }
  const int h = lane >> 2;
  const float a = ex[e * 8 + h] / den[(long)d * 8 + h];
  const int j0 = lane * 4;
  float4 xa = *(const float4*)(xl + (long)s * 128 + j0);
  float* o = out + (long)d * 128 + j0;
  atomicAdd(o + 0, xa.x * a);
  atomicAdd(o + 1, xa.y * a);
  atomicAdd(o + 2, xa.z * a);
  atomicAdd(o + 3, xa.w * a);
}

// ============ node post: +bias, elu, layernorm(128), +residual ==============
__global__ void node_post(const float* __restrict__ acc,
                          const float* __restrict__ bias,
                          const float* __restrict__ g,
                          const float* __restrict__ be,
                          const float* __restrict__ xin,
                          float* __restrict__ xout, int n) {
  const long node = (long)blockIdx.x * WPB + (threadIdx.x >> 5);
  if (node >= n) return;
  const int lane = threadIdx.x & 31;
  const int j0 = lane * 4;
  float4 v4 = *(const float4*)(acc + node * 128 + j0);
  float4 b4 = *(const float4*)(bias + j0);
  float h[4];
  float s = 0.0f, sq = 0.0f;
#pragma unroll
  for (int i = 0; i < 4; ++i) {
    float v = ((const float*)&v4)[i] + ((const float*)&b4)[i];
    v = v > 0.0f ? v : (__expf(v) - 1.0f);  // elu
    h[i] = v; s += v; sq += v * v;
  }
#pragma unroll
  for (int m = 1; m < 32; m <<= 1) {
    s += __shfl_xor(s, m, 32);
    sq += __shfl_xor(sq, m, 32);
  }
  const float mean = s * (1.0f / 128.0f);
  const float var = sq * (1.0f / 128.0f) - mean * mean;
  const float rstd = rsqrtf(var + 1e-5f);
  float4 g4 = *(const float4*)(g + j0);
  float4 e4 = *(const float4*)(be + j0);
  float4 x4 = *(const float4*)(xin + node * 128 + j0);
  float4 o;
#pragma unroll
  for (int i = 0; i < 4; ++i)
    ((float*)&o)[i] = (h[i] - mean) * rstd * ((const float*)&g4)[i] +
                      ((const float*)&e4)[i] + ((const float*)&x4)[i];
  *(float4*)(xout + node * 128 + j0) = o;
}

// ======== layer 2: head-mean, +bias, elu, layernorm(16), no residual =======
__global__ void node_post_mean(const float* __restrict__ acc,
                               const float* __restrict__ bias,
                               const float* __restrict__ g,
                               const float* __restrict__ be,
                               float* __restrict__ h2, int n) {
  long node = (long)blockIdx.x * blockDim.x + threadIdx.x;
  if (node >= n) return;
  float t[16];
  float s = 0.0f, sq = 0.0f;
#pragma unroll
  for (int c = 0; c < 16; ++c) {
    float v = 0.0f;
#pragma unroll
    for (int hh = 0; hh < 8; ++hh) v += acc[node * 128 + hh * 16 + c];
    v = v * 0.125f + bias[c];
    v = v > 0.0f ? v : (__expf(v) - 1.0f);
    t[c] = v; s += v; sq += v * v;
  }
  const float mean = s * (1.0f / 16.0f);
  const float var = sq * (1.0f / 16.0f) - mean * mean;
  const float rstd = rsqrtf(var + 1e-5f);
#pragma unroll
  for (int c = 0; c < 16; ++c)
    h2[node * 16 + c] = (t[c] - mean) * rstd * g[c] + be[c];
}

// ============================ launcher ======================================
extern "C" void kernel_launch(void* const* d_in, const int* in_sizes, int n_in,
                              void* d_out, int out_size, void* d_ws,
                              size_t ws_size, hipStream_t stream) {
  const int N = in_sizes[0] / 128;
  const int E = in_sizes[1] / 2;
  const int Etot = E + N;

  const float* x = (const float*)d_in[0];
  const int* src = (const int*)d_in[1];
  const int* dst = src + E;
  const float* ea = (const float*)d_in[2];
  const float* Wout = (const float*)d_in[30];
  const float* bout = (const float*)d_in[31];

  // workspace carve-out (~137 MB)
  float* ws = (float*)d_ws;
  size_t off = 0;
  float* xl    = ws + off; off += (size_t)N * 128;
  float* xr    = ws + off; off += (size_t)N * 128;
  float* accb  = ws + off; off += (size_t)N * 128;
  float* xbuf  = ws + off; off += (size_t)N * 128;
  float* alpha = ws + off; off += (size_t)Etot * 8;
  float* amax  = ws + off; off += (size_t)N * 8;  // uint keys then floats
  float* den   = ws + off; off += (size_t)N * 8;
  float* deg   = ws + off; off += (size_t)N;      // deg,easum adjacent
  float* easum = ws + off; off += (size_t)N * 4;  // becomes loop_ea
  float* h2    = ws + off; off += (size_t)N * 16;

  auto cdiv = [](long a, long b) { return (int)((a + b - 1) / b); };

  // self-loop edge attr = mean of incoming edge attrs per dst
  zero_f32<<<cdiv((long)N * 5, 256), 256, 0, stream>>>(deg, (long)N * 5);
  selfloop_accum<<<cdiv(E, 256), 256, 0, stream>>>(dst, ea, deg, easum, E);
  selfloop_final<<<cdiv(N, 256), 256, 0, stream>>>(deg, easum, N);

  const int edge_blocks = cdiv(Etot, WPB);
  const int node_blocks = cdiv(N, WPB);

  for (int l = 0; l < 3; ++l) {
    const int base = 3 + 9 * l;
    const float* Wl  = (const float*)d_in[base + 0];
    const float* bl  = (const float*)d_in[base + 1];
    const float* Wr  = (const float*)d_in[base + 2];
    const float* br  = (const float*)d_in[base + 3];
    const float* We  = (const float*)d_in[base + 4];
    const float* att = (const float*)d_in[base + 5];
    const float* bb  = (const float*)d_in[base + 6];
    const float* gg  = (const float*)d_in[base + 7];
    const float* be  = (const float*)d_in[base + 8];
    const float* xin = (l == 0) ? x : xbuf;

    gemm_wmma_f32<128><<<N / 16, 256, 0, stream>>>(xin, Wl, bl, xl);
    gemm_wmma_f32<128><<<N / 16, 256, 0, stream>>>(xin, Wr, br, xr);

    zero_f32<<<cdiv((long)N * 16, 256), 256, 0, stream>>>(amax, (long)N * 16);
    zero_f32<<<cdiv((long)N * 128, 256), 256, 0, stream>>>(accb, (long)N * 128);

    edge_alpha<<<edge_blocks, 256, 0, stream>>>(
        xl, xr, src, dst, ea, easum, We, att, alpha, (unsigned int*)amax, E,
        Etot);
    decode_amax<<<cdiv((long)N * 8, 256), 256, 0, stream>>>(
        (unsigned int*)amax, (long)N * 8);
    edge_exp<<<cdiv((long)Etot * 8, 256), 256, 0, stream>>>(
        alpha, dst, amax, den, E, (long)Etot * 8);
    edge_scatter<<<edge_blocks, 256, 0, stream>>>(xl, alpha, src, dst, den,
                                                  accb, E, Etot);

    if (l < 2)
      node_post<<<node_blocks, 256, 0, stream>>>(accb, bb, gg, be, xin, xbuf,
                                                 N);
    else
      node_post_mean<<<cdiv(N, 256), 256, 0, stream>>>(accb, bb, gg, be, h2,
                                                       N);
  }

  // out = h2[N,16] @ Wout[16,128] + bout
  gemm_wmma_f32<16><<<N / 16, 256, 0, stream>>>(h2, Wout, bout,
                                                (float*)d_out);
}